// MessagePassing_56530359550245
// MI455X (gfx1250) — compile-verified
//
#include <hip/hip_runtime.h>
#include <hip/hip_bf16.h>

typedef float v2f __attribute__((ext_vector_type(2)));
typedef float v8f __attribute__((ext_vector_type(8)));

#define D 128

// ---------------------------------------------------------------------------
// Kernel 1: zero a float buffer (grid-stride).
// ---------------------------------------------------------------------------
__global__ void mp_zero_kernel(float* __restrict__ p, long n) {
    long i = (long)blockIdx.x * blockDim.x + threadIdx.x;
    long stride = (long)gridDim.x * blockDim.x;
    for (; i < n; i += stride) p[i] = 0.0f;
}

// ---------------------------------------------------------------------------
// Kernel 2: per-type incoming-degree accumulation.
// adj is [L, E, 2] flattened; global edge index eg = l*E + e; tgt = adj[2*eg+1].
// ---------------------------------------------------------------------------
__global__ void mp_degree_kernel(const int* __restrict__ adj,
                                 float* __restrict__ deg,
                                 int n_nodes, int n_edges, int n_types) {
    long total = (long)n_types * n_edges;
    long i = (long)blockIdx.x * blockDim.x + threadIdx.x;
    if (i >= total) return;
    int l   = (int)(i / n_edges);
    int tgt = adj[2 * i + 1];
    atomicAdd(&deg[(size_t)l * n_nodes + tgt], 1.0f);
}

// ---------------------------------------------------------------------------
// Kernel 3: Y[l] = X @ W[l] via V_WMMA_F32_16X16X4_F32.
// One wave computes a 16x16 output tile; 8 waves/block cover columns 0..127 of
// one 16-row stripe. grid = (ceil(N/16), n_types).
//
// ISA lane layout for 32-bit 16x4 A / 4x16 B (wave32):
//   lanes 0-15  : K = k+0 (VGPR0), k+1 (VGPR1), row/col = lane
//   lanes 16-31 : K = k+2 (VGPR0), k+3 (VGPR1), row/col = lane-16
// C/D 16x16 f32: VGPR r -> (M=r, N=lane) for lanes 0-15, (M=r+8, N=lane-16).
// ---------------------------------------------------------------------------
__global__ void mp_gemm_wmma_f32(const float* __restrict__ X,
                                 const float* __restrict__ W,
                                 float* __restrict__ Y,
                                 int n_nodes) {
    const int l    = blockIdx.y;
    const int m0   = blockIdx.x * 16;
    const int wave = threadIdx.x >> 5;
    const int lane = threadIdx.x & 31;
    const int n0   = wave * 16;          // column tile base (8 waves * 16 = 128)
    const int half = lane >> 4;          // 0: K+{0,1}, 1: K+{2,3}
    const int idx  = lane & 15;          // A row within tile / B column within tile
    const int koff = half * 2;

    const float* __restrict__ Wl = W + (size_t)l * D * D;
    float* __restrict__ Yl       = Y + (size_t)l * n_nodes * D;

    // Clamp A row so every lane stays active (EXEC must be all-1s for WMMA).
    int arow_i = m0 + idx;
    if (arow_i >= n_nodes) arow_i = n_nodes - 1;
    const float* __restrict__ arow = X + (size_t)arow_i * D + koff;
    const float* __restrict__ bptr = Wl + (size_t)koff * D + (n0 + idx);

    v8f acc = {};
#pragma unroll 4
    for (int k = 0; k < D; k += 4) {
        v2f a;
        a.x = arow[k];
        a.y = arow[k + 1];
        v2f b;
        b.x = bptr[(size_t)k * D];
        b.y = bptr[(size_t)(k + 1) * D];
        acc = __builtin_amdgcn_wmma_f32_16x16x4_f32(
            /*neg_a=*/false, a, /*neg_b=*/false, b,
            /*c_mod=*/(short)0, acc, /*reuse_a=*/false, /*reuse_b=*/false);
    }

    // Store D: 8 rows per lane at stride D. Tile-in-range check is wave-uniform
    // -> single scalar branch; the fast path (always taken when N % 16 == 0)
    // has no per-row EXEC manipulation.
    const int mbase = m0 + half * 8;
    float* __restrict__ yout = Yl + (size_t)mbase * D + (n0 + idx);
    if (m0 + 16 <= n_nodes) {
#pragma unroll
        for (int r = 0; r < 8; ++r) yout[(size_t)r * D] = acc[r];
    } else {
#pragma unroll
        for (int r = 0; r < 8; ++r) {
            if (mbase + r < n_nodes) yout[(size_t)r * D] = acc[r];
        }
    }
}

// ---------------------------------------------------------------------------
// Kernel 4: edge scatter. One wave per edge: each of 32 lanes handles a
// contiguous float4 of the 128-wide row. Y[l][src] and out[tgt] live in L2
// (76.8 MB + 25.6 MB < 192 MB), so the random traffic stays on-chip.
// ---------------------------------------------------------------------------
__global__ void mp_scatter_kernel(const int* __restrict__ adj,
                                  const float* __restrict__ Y,
                                  const float* __restrict__ deg,
                                  float* __restrict__ out,
                                  int n_nodes, int n_edges, int n_types) {
    const int wave = threadIdx.x >> 5;
    const int lane = threadIdx.x & 31;
    const long eg  = (long)blockIdx.x * (blockDim.x >> 5) + wave; // global edge idx
    const long total = (long)n_types * n_edges;
    if (eg >= total) return;

    const int l   = (int)(eg / n_edges);
    const int src = adj[2 * eg];
    const int tgt = adj[2 * eg + 1];

    const float dv = deg[(size_t)l * n_nodes + tgt];
    const float s  = rsqrtf(fmaxf(dv, 1.0f));

    const float4* __restrict__ yrow =
        (const float4*)(Y + ((size_t)l * n_nodes + src) * D);
    float4 v = yrow[lane];

    float* __restrict__ o = out + (size_t)tgt * D + lane * 4;
    atomicAdd(o + 0, v.x * s);
    atomicAdd(o + 1, v.y * s);
    atomicAdd(o + 2, v.z * s);
    atomicAdd(o + 3, v.w * s);
}

// ---------------------------------------------------------------------------
// Launcher. Inputs: d_in[0]=node_embeddings [N,128] f32,
//                   d_in[1]=adjacency [L,E,2] i32,
//                   d_in[2]=W [L,128,128] f32.
// Workspace layout: deg[L*N] floats, then Y[L*N*128] floats (~77.4 MB).
// ---------------------------------------------------------------------------
extern "C" void kernel_launch(void* const* d_in, const int* in_sizes, int n_in,
                              void* d_out, int out_size, void* d_ws, size_t ws_size,
                              hipStream_t stream) {
    const float* X   = (const float*)d_in[0];
    const int*   adj = (const int*)d_in[1];
    const float* W   = (const float*)d_in[2];
    float*       out = (float*)d_out;

    const int n_nodes = in_sizes[0] / D;
    const int n_types = in_sizes[2] / (D * D);
    const int n_edges = in_sizes[1] / (n_types * 2);

    float* deg = (float*)d_ws;                       // n_types * n_nodes floats
    float* Y   = deg + (size_t)n_types * n_nodes;    // n_types * n_nodes * D floats

    // 1) Zero output and degree counters (fresh every call -> deterministic).
    {
        long n_out = (long)n_nodes * D;
        int blocks = (int)((n_out + 255) / 256);
        if (blocks > 4096) blocks = 4096;
        mp_zero_kernel<<<blocks, 256, 0, stream>>>(out, n_out);

        long n_deg = (long)n_types * n_nodes;
        int dblocks = (int)((n_deg + 255) / 256);
        if (dblocks > 4096) dblocks = 4096;
        mp_zero_kernel<<<dblocks, 256, 0, stream>>>(deg, n_deg);
    }

    // 2) Per-type incoming degrees.
    {
        long total = (long)n_types * n_edges;
        int blocks = (int)((total + 255) / 256);
        mp_degree_kernel<<<blocks, 256, 0, stream>>>(adj, deg, n_nodes, n_edges,
                                                     n_types);
    }

    // 3) Y[l] = X @ W[l] with f32 WMMA. 8 waves/block -> 16 rows x 128 cols.
    {
        dim3 grid((n_nodes + 15) / 16, n_types);
        mp_gemm_wmma_f32<<<grid, 256, 0, stream>>>(X, W, Y, n_nodes);
    }

    // 4) Gather-scale-scatter over edges: one wave per edge, 8 edges/block.
    {
        long total = (long)n_types * n_edges;
        int blocks = (int)((total + 7) / 8);
        mp_scatter_kernel<<<blocks, 256, 0, stream>>>(adj, Y, deg, out, n_nodes,
                                                      n_edges, n_types);
    }
}